// DeltaNetBlock_4063039062141
// MI455X (gfx1250) — compile-verified
//
#include <hip/hip_runtime.h>
#include <hip/hip_bf16.h>
#include <math.h>

// ---------------------------------------------------------------------------
// DeltaNet block, algebraically collapsed (last_state == 0):
//   kp = x@Wk.T+bk ; vp = x@Wv.T+bv                (WMMA f32 GEMM)
//   kc = l2norm(silu(conv3(kp))) ; vc = silu(conv3(vp))   (WMMA f32 GEMMs)
//   beta = sigmoid(x@Wb.T+bb)                      (wave GEMV)
//   P = vc @ kc  (384x384x384, WMMA f32)
//   n[a,b] = d*rsqrt(d*d+eps), d = beta[b]*P[b,a]
//   w[o] = sum_c g[c]*Wo[o,c]                      (wave GEMV)
//   out[a,b,o] = n[a,b]*w[o] + bo[o]               (226MB nontemporal stream)
// ---------------------------------------------------------------------------

typedef __attribute__((ext_vector_type(2))) float v2f;
typedef __attribute__((ext_vector_type(4))) float v4f;
typedef __attribute__((ext_vector_type(8))) float v8f;

#define DD 384
#define TILES_1D 24          // 384 / 16
#define N_TILES (TILES_1D * TILES_1D)

// ---- y[s,o] = sum_i x[s,i]*W[o,i] + b[o] ; one wave per 16x16 tile --------
__global__ __launch_bounds__(32)
void proj_wmma(const float* __restrict__ x, const float* __restrict__ W,
               const float* __restrict__ bias, float* __restrict__ y) {
  const int tile = blockIdx.x;
  const int tm = (tile / TILES_1D) * 16;   // s-tile
  const int tn = (tile % TILES_1D) * 16;   // o-tile
  const int lane = threadIdx.x & 31;
  const int half = lane >> 4;              // selects K pair {2h, 2h+1}
  const int lid  = lane & 15;
  v8f acc = {};
  const float* arow = x + (size_t)(tm + lid) * DD;   // A: x[s, k]
  const float* brow = W + (size_t)(tn + lid) * DD;   // B: W^T[k,o] = W[o,k]
  for (int k0 = 0; k0 < DD; k0 += 4) {
    const int kk = k0 + 2 * half;
    v2f a; a.x = arow[kk]; a.y = arow[kk + 1];
    v2f b; b.x = brow[kk]; b.y = brow[kk + 1];
    acc = __builtin_amdgcn_wmma_f32_16x16x4_f32(false, a, false, b,
                                                (short)0, acc, false, false);
  }
  const int o = tn + lid;
  const float bv = bias[o];
#pragma unroll
  for (int r = 0; r < 8; ++r) {
    const int s = tm + r + 8 * half;       // C/D layout: VGPR r -> M=r+8*half
    y[(size_t)s * DD + o] = acc[r] + bv;
  }
}

// ---- out[o,l] = silu( sum_t sum_c Wc[o,c,t]*in[c,l+t-1] + b[o] ) ----------
__global__ __launch_bounds__(32)
void conv_silu_wmma(const float* __restrict__ in,   // (C=384, L=384)
                    const float* __restrict__ Wc,   // (O, C, 3)
                    const float* __restrict__ bias,
                    float* __restrict__ out) {      // (O, L)
  const int tile = blockIdx.x;
  const int tm = (tile / TILES_1D) * 16;   // o-tile
  const int tn = (tile % TILES_1D) * 16;   // l-tile
  const int lane = threadIdx.x & 31;
  const int half = lane >> 4;
  const int lid  = lane & 15;
  v8f acc = {};
  const int orow = tm + lid;
  const int lcol = tn + lid;
  for (int t = 0; t < 3; ++t) {
    const int col = lcol + t - 1;                     // zero-pad 'same'
    const bool ok = (col >= 0) && (col < DD);
    for (int k0 = 0; k0 < DD; k0 += 4) {
      const int c0 = k0 + 2 * half;
      v2f a;                                          // A[m,k]=Wc[o, c, t]
      a.x = Wc[((size_t)orow * DD + c0) * 3 + t];
      a.y = Wc[((size_t)orow * DD + c0 + 1) * 3 + t];
      v2f b;                                          // B[k,n]=in[c, l+t-1]
      b.x = ok ? in[(size_t)c0 * DD + col] : 0.0f;
      b.y = ok ? in[(size_t)(c0 + 1) * DD + col] : 0.0f;
      acc = __builtin_amdgcn_wmma_f32_16x16x4_f32(false, a, false, b,
                                                  (short)0, acc, false, false);
    }
  }
  const int l = tn + lid;
#pragma unroll
  for (int r = 0; r < 8; ++r) {
    const int o = tm + r + 8 * half;
    const float yv = acc[r] + bias[o];
    const float sv = yv / (1.0f + __expf(-yv));       // SiLU
    out[(size_t)o * DD + l] = sv;
  }
}

// ---- normalize each of 384 rows (len 384) to unit L2 (eps 1e-12) ----------
__global__ __launch_bounds__(32)
void l2norm_rows(float* __restrict__ a) {
  const int row = blockIdx.x;
  const int lane = threadIdx.x & 31;
  float s = 0.0f;
  for (int j = lane; j < DD; j += 32) {
    const float v = a[(size_t)row * DD + j];
    s += v * v;
  }
  for (int off = 16; off > 0; off >>= 1) s += __shfl_xor(s, off, 32);
  const float inv = 1.0f / fmaxf(sqrtf(s), 1e-12f);
  for (int j = lane; j < DD; j += 32) a[(size_t)row * DD + j] *= inv;
}

// ---- beta[s] = sigmoid( x[s,:].Wb + bb ) ----------------------------------
__global__ __launch_bounds__(32)
void beta_gemv(const float* __restrict__ x, const float* __restrict__ Wb,
               const float* __restrict__ bb, float* __restrict__ beta) {
  const int s = blockIdx.x;
  const int lane = threadIdx.x & 31;
  float acc = 0.0f;
  for (int i = lane; i < DD; i += 32) acc += x[(size_t)s * DD + i] * Wb[i];
  for (int off = 16; off > 0; off >>= 1) acc += __shfl_xor(acc, off, 32);
  if (lane == 0) beta[s] = 1.0f / (1.0f + __expf(-(acc + bb[0])));
}

// ---- w[o] = sum_c g[c]*Wo[o,c] --------------------------------------------
__global__ __launch_bounds__(32)
void wvec_gemv(const float* __restrict__ g, const float* __restrict__ Wo,
               float* __restrict__ w) {
  const int o = blockIdx.x;
  const int lane = threadIdx.x & 31;
  float acc = 0.0f;
  for (int c = lane; c < DD; c += 32) acc += g[c] * Wo[(size_t)o * DD + c];
  for (int off = 16; off > 0; off >>= 1) acc += __shfl_xor(acc, off, 32);
  if (lane == 0) w[o] = acc;
}

// ---- P = V@K ; n[s,m] = d*rsqrt(d*d+eps), d = beta[m]*P[m,s] --------------
__global__ __launch_bounds__(32)
void delta_norm_wmma(const float* __restrict__ V,    // vc (384x384)
                     const float* __restrict__ K,    // kc (384x384)
                     const float* __restrict__ beta,
                     float* __restrict__ nmat) {     // n stored at [a*384+b]
  const int tile = blockIdx.x;
  const int tm = (tile / TILES_1D) * 16;   // m-tile
  const int tn = (tile % TILES_1D) * 16;   // s-tile
  const int lane = threadIdx.x & 31;
  const int half = lane >> 4;
  const int lid  = lane & 15;
  v8f acc = {};
  const float* arow = V + (size_t)(tm + lid) * DD;
  for (int k0 = 0; k0 < DD; k0 += 4) {
    const int j0 = k0 + 2 * half;
    v2f a; a.x = arow[j0]; a.y = arow[j0 + 1];
    v2f b;                                  // B[k,n] = K[j, s]
    b.x = K[(size_t)j0 * DD + tn + lid];
    b.y = K[(size_t)(j0 + 1) * DD + tn + lid];
    acc = __builtin_amdgcn_wmma_f32_16x16x4_f32(false, a, false, b,
                                                (short)0, acc, false, false);
  }
  const int sIdx = tn + lid;                // a-index of n
#pragma unroll
  for (int r = 0; r < 8; ++r) {
    const int m = tm + r + 8 * half;        // b-index of n
    const float d = beta[m] * acc[r];
    const float val = d * rsqrtf(d * d + 1.1920929e-07f);  // RMSNorm collapse
    nmat[(size_t)sIdx * DD + m] = val;
  }
}

// ---- out[a,b,o] = n[a,b]*w[o] + bo[o] : 226 MB nontemporal stream ---------
__global__ __launch_bounds__(256)
void broadcast_out(const float* __restrict__ nmat, const float* __restrict__ w,
                   const float* __restrict__ bo, float* __restrict__ out) {
  const unsigned idx = blockIdx.x * 256u + threadIdx.x;  // one float4 each
  const unsigned flat = idx * 4u;                        // < 384^3 < 2^31
  const unsigned ab = flat / DD;
  const unsigned o  = flat % DD;                         // multiple of 4
  const float nv = nmat[ab];
  const v4f wv = *(const v4f*)(w + o);
  const v4f bv = *(const v4f*)(bo + o);
  v4f res;
  res.x = nv * wv.x + bv.x;
  res.y = nv * wv.y + bv.y;
  res.z = nv * wv.z + bv.z;
  res.w = nv * wv.w + bv.w;
  __builtin_nontemporal_store(res, (v4f*)(out + (size_t)flat));
}

// ---------------------------------------------------------------------------
extern "C" void kernel_launch(void* const* d_in, const int* in_sizes, int n_in,
                              void* d_out, int out_size, void* d_ws, size_t ws_size,
                              hipStream_t stream) {
  const float* x   = (const float*)d_in[0];
  const float* Wk  = (const float*)d_in[1];
  const float* bk  = (const float*)d_in[2];
  // d_in[3], d_in[4]: Wq/bq — q is dead code in the reference, skipped.
  const float* Wv  = (const float*)d_in[5];
  const float* bv  = (const float*)d_in[6];
  const float* Wkc = (const float*)d_in[7];
  const float* bkc = (const float*)d_in[8];
  // d_in[9], d_in[10]: Wqc/bqc — dead, skipped.
  const float* Wvc = (const float*)d_in[11];
  const float* bvc = (const float*)d_in[12];
  const float* Wb  = (const float*)d_in[13];
  const float* bb  = (const float*)d_in[14];
  const float* g   = (const float*)d_in[15];
  const float* Wo  = (const float*)d_in[16];
  const float* bo  = (const float*)d_in[17];
  float* out = (float*)d_out;

  const size_t M = (size_t)DD * DD;                 // 147456 floats
  float* ws   = (float*)d_ws;
  float* kp   = ws;                                 // k projection
  float* vp   = kp + M;                             // v projection
  float* kc   = vp + M;                             // k after conv/silu/norm
  float* vc   = kc + M;                             // v after conv/silu
  float* nmat = vc + M;                             // n[a,b]
  float* beta = nmat + M;                           // 384
  float* wvec = beta + DD;                          // 384

  proj_wmma<<<N_TILES, 32, 0, stream>>>(x, Wk, bk, kp);
  proj_wmma<<<N_TILES, 32, 0, stream>>>(x, Wv, bv, vp);
  conv_silu_wmma<<<N_TILES, 32, 0, stream>>>(kp, Wkc, bkc, kc);
  conv_silu_wmma<<<N_TILES, 32, 0, stream>>>(vp, Wvc, bvc, vc);
  l2norm_rows<<<DD, 32, 0, stream>>>(kc);
  beta_gemv<<<DD, 32, 0, stream>>>(x, Wb, bb, beta);
  delta_norm_wmma<<<N_TILES, 32, 0, stream>>>(vc, kc, beta, nmat);
  wvec_gemv<<<DD, 32, 0, stream>>>(g, Wo, wvec);

  // 384^3 floats = 14,155,776 float4 = 55,296 blocks of 256 threads, exact.
  broadcast_out<<<(DD * DD * DD) / (4 * 256), 256, 0, stream>>>(nmat, wvec, bo, out);
}